// FuzzyPooling_4698694222169
// MI455X (gfx1250) — compile-verified
//
#include <hip/hip_runtime.h>
#include <cstdint>

// FuzzyPooling: out[b,c,oh,ow] = mean_{2x2}( x*exp(-x^2/2) ), x:(32,64,224,224) f32
//
// Memory-bound streaming kernel (~514 MB moved, ~22us floor @ 23.3 TB/s).
// CDNA5 path: GLOBAL_LOAD_ASYNC_TO_LDS_B128 (GVS form, NT hint) double-buffered
// pipeline tracked by ASYNCcnt; ds_load_b128 consume; v_exp_f32 math; NT stores.
//
// Work partition: 12,845,056 output pairs = 1568 blocks x 256 thr x 32 iters.
// TSTRIDE = 401408 = 56*112*64, so per thread owp/oh are loop-invariant and
// only bc advances (+64/iter): input addr += 12,845,056 B, output += 802,816
// floats... (64*112*112) per iteration -> zero div/mod inside the loop.

typedef float v2f __attribute__((ext_vector_type(2)));

#define NTHR    256
#define NBLK    1568
#define ITERS   32
#define TSTRIDE (NBLK * NTHR)              // 401408
#define IN_STEP  (64u * 224u * 224u * 4u)  // bytes per iteration (12,845,056)
#define OUT_STEP (64u * 112u * 112u)       // floats per iteration (802,816)

__device__ __forceinline__ float fuzzy(float v) {
    // v * exp(-v*v/2) == v * exp2(v*v * (-0.5 * log2(e)))
    return v * __builtin_amdgcn_exp2f(v * v * -0.7213475204444817f);
}

__global__ __launch_bounds__(NTHR, 2)
void fuzzy_pool_kernel(const float* __restrict__ x, float* __restrict__ out)
{
    // [stage][slot][lane] : slot-major so ds_load_b128 has 16B lane stride (conflict-free)
    __shared__ float4 buf[2][2][NTHR];

    const uint32_t tid = threadIdx.x;
    // Low 32 bits of a generic pointer to LDS == wave-relative LDS byte offset.
    const uint32_t ldsLane = (uint32_t)(uintptr_t)(&buf[0][0][0]) + tid * 16u;

    // Decompose pair id ONCE; owp/oh are invariant across the iteration loop.
    const uint32_t p   = blockIdx.x * NTHR + tid;
    const uint32_t owp = p % 56u;
    const uint32_t t   = p / 56u;
    const uint32_t oh  = t % 112u;
    const uint32_t bc  = t / 112u;

    // Input byte offset of the float4 at (bc, 2*oh, 4*owp); row pitch 896 B.
    uint32_t inOff = ((bc * 224u + oh * 2u) * 224u + owp * 4u) * 4u;
    float*   op    = out + ((bc * 112u + oh) * 112u + owp * 2u);

    auto issue = [&](uint32_t byteOff, uint32_t stage) {
        uint32_t la = ldsLane + stage * 8192u;
        asm volatile(
            "global_load_async_to_lds_b128 %0, %2, %3 th:TH_LOAD_NT\n\t"
            "global_load_async_to_lds_b128 %1, %2, %3 offset:896 th:TH_LOAD_NT"
            :
            : "v"(la), "v"(la + 4096u), "v"(byteOff), "s"(x)
            : "memory");
    };

    issue(inOff, 0u);

    for (int i = 0; i < ITERS; ++i) {
        const uint32_t s = (uint32_t)i & 1u;
        if (i + 1 < ITERS) {
            issue(inOff + IN_STEP, s ^ 1u);
            // 4 async loads in flight max; <=2 left guarantees stage s landed
            asm volatile("s_wait_asynccnt 0x2" ::: "memory");
        } else {
            asm volatile("s_wait_asynccnt 0x0" ::: "memory");
        }

        float4 r0 = buf[s][0][tid];   // row 2*oh
        float4 r1 = buf[s][1][tid];   // row 2*oh+1

        float o0 = 0.25f * (fuzzy(r0.x) + fuzzy(r0.y) + fuzzy(r1.x) + fuzzy(r1.y));
        float o1 = 0.25f * (fuzzy(r0.z) + fuzzy(r0.w) + fuzzy(r1.z) + fuzzy(r1.w));

        v2f o; o.x = o0; o.y = o1;
        __builtin_nontemporal_store(o, (v2f*)op);

        inOff += IN_STEP;
        op    += OUT_STEP;
    }
}

extern "C" void kernel_launch(void* const* d_in, const int* in_sizes, int n_in,
                              void* d_out, int out_size, void* d_ws, size_t ws_size,
                              hipStream_t stream) {
    const float* x = (const float*)d_in[0];
    float* out = (float*)d_out;
    fuzzy_pool_kernel<<<NBLK, NTHR, 0, stream>>>(x, out);
}